// FloatingRegionScore_56822417326793
// MI455X (gfx1250) — compile-verified
//
#include <hip/hip_runtime.h>
#include <cstdint>

#define HH 512
#define WW 1024
#define HWPX (HH * WW)
#define NC 19
#define RAD 16
#define KS 33
// 1/ln(19)
#define INV_LOGC 0.339622536897904f

typedef float v2f __attribute__((ext_vector_type(2)));
typedef float v8f __attribute__((ext_vector_type(8)));

// ---------------------------------------------------------------------------
// K1: per-pixel softmax entropy (already divided by ln C) + argmax class.
// One coalesced pass over the 40 MB of logits (the only mandatory HBM read).
// ---------------------------------------------------------------------------
__global__ void k1_entropy_argmax(const float* __restrict__ logit,
                                  float* __restrict__ ent,
                                  unsigned char* __restrict__ pred) {
    int i = blockIdx.x * blockDim.x + threadIdx.x;
    if (i >= HWPX) return;

    float v[NC];
#pragma unroll
    for (int c = 0; c < NC; ++c) v[c] = logit[c * HWPX + i];

    float m = v[0];
    int am = 0;
#pragma unroll
    for (int c = 1; c < NC; ++c) {
        if (v[c] > m) { m = v[c]; am = c; }
    }

    float s = 0.f;
#pragma unroll
    for (int c = 0; c < NC; ++c) { v[c] = __expf(v[c] - m); s += v[c]; }

    float inv = 1.f / s;
    float e = 0.f;
#pragma unroll
    for (int c = 0; c < NC; ++c) {
        float p = v[c] * inv;
        e -= p * __logf(p + 1e-6f);
    }
    ent[i] = e * INV_LOGC;
    pred[i] = (unsigned char)am;
}

// ---------------------------------------------------------------------------
// K2: horizontal 33-tap sums via LDS staging. Class histogram packed as
// 4 x u32 with 6-bit fields (5 classes per word, counts <= 33 < 64, so packed
// adds never carry across fields). 4 adds per tap instead of 19 compares.
// One workgroup = one 256-pixel row chunk (+16 halo each side).
// ---------------------------------------------------------------------------
__global__ void k2_hpass(const float* __restrict__ ent,
                         const unsigned char* __restrict__ pred,
                         float* __restrict__ ent_h,
                         uint4* __restrict__ cnt_h) {
    __shared__ float es[256 + 2 * RAD];
    __shared__ uint4 ps[256 + 2 * RAD];

    int bid = blockIdx.x;
    int y = bid >> 2;            // 4 chunks per row (W = 1024)
    int x0 = (bid & 3) << 8;
    int tid = threadIdx.x;

    for (int j = tid; j < 256 + 2 * RAD; j += 256) {
        int xx = x0 - RAD + j;
        float e = 0.f;
        uint4 q = make_uint4(0u, 0u, 0u, 0u);
        if (xx >= 0 && xx < WW) {
            int idx = y * WW + xx;
            e = ent[idx];
            int c = pred[idx];
            unsigned bit = 1u << (6 * (c % 5));
            int w = c / 5;
            if (w == 0) q.x = bit;
            else if (w == 1) q.y = bit;
            else if (w == 2) q.z = bit;
            else q.w = bit;
        }
        es[j] = e;
        ps[j] = q;
    }
    __syncthreads();

    float s = 0.f;
    unsigned a0 = 0, a1 = 0, a2 = 0, a3 = 0;
#pragma unroll 11
    for (int d = 0; d < KS; ++d) {
        s += es[tid + d];
        uint4 q = ps[tid + d];
        a0 += q.x; a1 += q.y; a2 += q.z; a3 += q.w;
    }
    int idx = y * WW + x0 + tid;
    ent_h[idx] = s;
    cnt_h[idx] = make_uint4(a0, a1, a2, a3);
}

// ---------------------------------------------------------------------------
// K3: vertical 33-tap box-sum of ent_h as a banded-ones GEMM on the CDNA5
// matrix pipe:  out(16x16) = Band(16x48) x in(48x16), done as 12 chained
// V_WMMA_F32_16X16X4_F32. One wave per 16x16 tile.
//
// Zero-padding is folded into the band matrix A (whole K-column zeroed for
// rows outside the image), so every B load uses a CLAMPED in-bounds address
// and is issued unconditionally: no exec-mask branching, all 24 loads are
// batched ahead of the 12 back-to-back WMMAs (one loadcnt wait total).
// f32 MACs -> numerically identical to scalar f32 summation (RNE).
// ---------------------------------------------------------------------------
__global__ void k3_vpass_wmma(const float* __restrict__ ent_h,
                              float* __restrict__ ent_v) {
    int lane = threadIdx.x & 31;
    int wave = threadIdx.x >> 5;
    int tile = blockIdx.x * 8 + wave;   // 2048 tiles = (512/16) x (1024/16)
    int ty = tile >> 6;                 // tile row (64 tiles per image row band)
    int tx = tile & 63;
    int y0 = ty << 4, x0 = tx << 4;

    int half = lane >> 4;   // 0: lanes 0-15, 1: lanes 16-31
    int mn = lane & 15;     // m for A-layout, n for B/D-layout

    // Preload all 24 B elements (clamped addresses, unconditional loads) and
    // synthesize the 24 A elements (band * row-validity) in registers.
    float bv[12][2];
    float av[12][2];
#pragma unroll
    for (int chunk = 0; chunk < 12; ++chunk) {
#pragma unroll
        for (int v = 0; v < 2; ++v) {
            int k = v + 2 * half;          // K index per 16x16x4 f32 VGPR layout
            int rr = chunk * 4 + k;        // window-relative input row, 0..47
            int gy = y0 - RAD + rr;
            int gyc = min(max(gy, 0), HH - 1);     // clamped, always in-bounds
            bv[chunk][v] = ent_h[gyc * WW + x0 + mn];
            // Band: output row m sums input rows rr in [m, m+32]; rows that
            // fall outside the image contribute a zero A-column instead.
            bool valid = (gy >= 0) & (gy < HH);
            av[chunk][v] = (valid && rr >= mn && rr <= mn + 32) ? 1.0f : 0.0f;
        }
    }

    v8f acc = {0.f, 0.f, 0.f, 0.f, 0.f, 0.f, 0.f, 0.f};
#pragma unroll
    for (int chunk = 0; chunk < 12; ++chunk) {
        v2f a = {av[chunk][0], av[chunk][1]};
        v2f b = {bv[chunk][0], bv[chunk][1]};
        acc = __builtin_amdgcn_wmma_f32_16x16x4_f32(
            false, a, false, b, (short)0, acc, false, false);
    }

#pragma unroll
    for (int r = 0; r < 8; ++r) {
        int gm = r + 8 * half;        // D layout: VGPR r -> rows r / r+8
        ent_v[(y0 + gm) * WW + x0 + mn] = acc[r];
    }
}

// ---------------------------------------------------------------------------
// K4: vertical running class-count (19 register counters, updated by +/- one
// packed row per step) + final impurity / uncertainty / score. `count` is the
// clipped-window pixel count, computed in closed form. Threads = columns
// (coalesced uint4 loads of cnt_h), each marching a 16-row strip.
// ---------------------------------------------------------------------------
#define CHUNK 16
__global__ void k4_vcount_final(const uint4* __restrict__ cnt_h,
                                const float* __restrict__ ent_v,
                                float* __restrict__ out) {
    int gid = blockIdx.x * blockDim.x + threadIdx.x;  // W * (H/CHUNK) threads
    int x = gid & (WW - 1);
    int ystart = (gid >> 10) * CHUNK;

    int cnt[NC];
#pragma unroll
    for (int c = 0; c < NC; ++c) cnt[c] = 0;

    // warm-up: rows [ystart-16, ystart+16] clipped to the image
    for (int yy = ystart - RAD; yy <= ystart + RAD; ++yy) {
        if (yy < 0 || yy >= HH) continue;
        uint4 q = cnt_h[yy * WW + x];
        unsigned sw[4] = {q.x, q.y, q.z, q.w};
#pragma unroll
        for (int c = 0; c < NC; ++c)
            cnt[c] += (int)((sw[c / 5] >> (6 * (c % 5))) & 63u);
    }

    int cx = min(x + RAD, WW - 1) - max(x - RAD, 0) + 1;

    for (int y = ystart; y < ystart + CHUNK; ++y) {
        int cy = min(y + RAD, HH - 1) - max(y - RAD, 0) + 1;
        float invc = 1.f / (float)(cx * cy);

        float imp = 0.f;
#pragma unroll
        for (int c = 0; c < NC; ++c) {
            float d = (float)cnt[c] * invc;
            imp -= d * __logf(d + 1e-6f);
        }
        imp *= INV_LOGC;

        int idx = y * WW + x;
        float pu = ent_v[idx] * invc;
        out[idx] = imp * pu;            // score
        out[HWPX + idx] = imp;          // region impurity
        out[2 * HWPX + idx] = pu;       // prediction uncertainty

        // slide window down one row
        int yin = y + RAD + 1, yout = y - RAD;
        if (yin < HH) {
            uint4 q = cnt_h[yin * WW + x];
            unsigned sw[4] = {q.x, q.y, q.z, q.w};
#pragma unroll
            for (int c = 0; c < NC; ++c)
                cnt[c] += (int)((sw[c / 5] >> (6 * (c % 5))) & 63u);
        }
        if (yout >= 0) {
            uint4 q = cnt_h[yout * WW + x];
            unsigned sw[4] = {q.x, q.y, q.z, q.w};
#pragma unroll
            for (int c = 0; c < NC; ++c)
                cnt[c] -= (int)((sw[c / 5] >> (6 * (c % 5))) & 63u);
        }
    }
}

// ---------------------------------------------------------------------------
// Launch: logits -> (ent, pred) -> horizontal sums -> WMMA vertical entropy
// sum -> vertical counts + final outputs. Workspace: 13*HW bytes of f32/u8
// intermediates + 16*HW bytes packed counts (~14.5 MB, L2 resident).
// ---------------------------------------------------------------------------
extern "C" void kernel_launch(void* const* d_in, const int* in_sizes, int n_in,
                              void* d_out, int out_size, void* d_ws, size_t ws_size,
                              hipStream_t stream) {
    (void)in_sizes; (void)n_in; (void)out_size; (void)ws_size;

    const float* logit = (const float*)d_in[0];
    float* out = (float*)d_out;

    char* ws = (char*)d_ws;
    float* ent            = (float*)(ws);                             // 2 MB
    float* ent_h          = (float*)(ws + (size_t)HWPX * 4);          // 2 MB
    float* ent_v          = (float*)(ws + (size_t)HWPX * 8);          // 2 MB
    unsigned char* pred   = (unsigned char*)(ws + (size_t)HWPX * 12); // 0.5 MB
    uint4* cnt_h          = (uint4*)(ws + (size_t)HWPX * 13);         // 8 MB (16B aligned)

    k1_entropy_argmax<<<HWPX / 256, 256, 0, stream>>>(logit, ent, pred);
    k2_hpass<<<HH * (WW / 256), 256, 0, stream>>>(ent, pred, ent_h, cnt_h);
    k3_vpass_wmma<<<(HH / 16) * (WW / 16) / 8, 256, 0, stream>>>(ent_h, ent_v);
    k4_vcount_final<<<WW * (HH / CHUNK) / 256, 256, 0, stream>>>(cnt_h, ent_v, out);
}